// SoftGraphCut_57251914055800
// MI455X (gfx1250) — compile-verified
//
#include <hip/hip_runtime.h>
#include <stdint.h>

#define H 512
#define W 512
#define PLANE (H * W)
#define NB 16
#define NC 3
#define NPL (NB * NC)              // 48 independent (b,c) planes
#define TILE_H 8
#define TILES_PP (H / TILE_H)      // 64 row-tiles per plane
#define GRID1 (NPL * TILES_PP)     // 3072 blocks
#define NTHREADS 256
#define MAX_ITERS 100
// eps = (1e-6 * 16*512*512*3)^2 = 12.582912^2
#define EPS_CONV 158.32967f
#define FMAXV 3.4028234663852886e38f

typedef float v2f   __attribute__((ext_vector_type(2)));
typedef float v8f   __attribute__((ext_vector_type(8)));
typedef unsigned int u32x4 __attribute__((ext_vector_type(4)));
typedef int   i32x4 __attribute__((ext_vector_type(4)));
typedef int   i32x8 __attribute__((ext_vector_type(8)));

// ---------------- helpers ----------------

__device__ __forceinline__ float nan_to_num_dev(float v) {
    if (__builtin_isnan(v)) return 0.f;
    if (__builtin_isinf(v)) return v > 0.f ? FMAXV : -FMAXV;
    return v;
}

// Exact f32 32-lane sum. Uses V_WMMA_F32_16X16X4_F32 as a fold engine:
// A[m][0] = partial(lane m), A[m][2] = partial(lane m+16) (ISA 16x4 A layout),
// B = ones  =>  D[m][*] = partial(m)+partial(m+16).  Sum of the 8 D VGPRs in
// lane L<16 covers lanes {0..7,16..23}; lanes>=16 cover {8..15,24..31}; one
// SWAPX16 ds_swizzle completes the reduction (all-lanes result).
__device__ __forceinline__ float waveReduceSum(float s) {
#if __has_builtin(__builtin_amdgcn_wmma_f32_16x16x4_f32)
    v2f a; a[0] = s;   a[1] = 0.f;
    v2f b; b[0] = 1.f; b[1] = 1.f;
    v8f c = {};
    v8f d = __builtin_amdgcn_wmma_f32_16x16x4_f32(false, a, false, b,
                                                  (short)0, c, false, false);
    float t = ((d[0] + d[1]) + (d[2] + d[3])) + ((d[4] + d[5]) + (d[6] + d[7]));
#else
    float t = s;
    t += __int_as_float(__builtin_amdgcn_ds_swizzle(__float_as_int(t), (0x01 << 10) | 0x1F));
    t += __int_as_float(__builtin_amdgcn_ds_swizzle(__float_as_int(t), (0x02 << 10) | 0x1F));
    t += __int_as_float(__builtin_amdgcn_ds_swizzle(__float_as_int(t), (0x04 << 10) | 0x1F));
    t += __int_as_float(__builtin_amdgcn_ds_swizzle(__float_as_int(t), (0x08 << 10) | 0x1F));
#endif
    t += __int_as_float(__builtin_amdgcn_ds_swizzle(__float_as_int(t), (0x10 << 10) | 0x1F));
    return t;
}

// Block (256-thread) sum; valid result on thread 0.
__device__ __forceinline__ float blockReduceSum(float s, float* red) {
    float w = waveReduceSum(s);
    int t = threadIdx.x;
    if ((t & 31) == 0) red[t >> 5] = w;
    __syncthreads();
    float tot = 0.f;
    if (t == 0) {
        #pragma unroll
        for (int i = 0; i < NTHREADS / 32; ++i) tot += red[i];
    }
    __syncthreads();
    return tot;
}

// ---- Tensor Data Mover: 2-D strided slab (global f32) -> LDS ----
// D# per CDNA5 ISA.  Issue from ONE wave only
// (TDM ignores EXEC); completion via TENSORcnt.
// This toolchain's builtin takes 6 args:
//   (uint32x4 g0, int32x8 g1, int32x4 g2, int32x4 g3, int32x8 extra, i32 cpol)
#if __has_builtin(__builtin_amdgcn_tensor_load_to_lds)
#define HAVE_TDM 1
__device__ __forceinline__ void tdm_load_2d_f32(const float* gsrc,
                                                unsigned lds_byte_off,
                                                unsigned width_elems,
                                                unsigned nrows,
                                                unsigned long long stride_elems) {
    unsigned long long ga = (unsigned long long)(uintptr_t)gsrc;
    u32x4 g0;
    g0[0] = 1u;                                          // count=1 (valid user D#)
    g0[1] = lds_byte_off;                                // lds_addr
    g0[2] = (unsigned)(ga & 0xFFFFFFFFu);                // global_addr[31:0]
    g0[3] = (unsigned)((ga >> 32) & 0x1FFFFFFu) | (2u << 30);  // addr[56:32] | type=2
    i32x8 g1;
    g1[0] = (int)(2u << 16);                             // data_size=2 -> 4 bytes
    g1[1] = (int)((width_elems & 0xFFFFu) << 16);        // tensor_dim0[15:0]
    g1[2] = (int)(((width_elems >> 16) & 0xFFFFu) | ((nrows & 0xFFFFu) << 16));   // dim0 hi | dim1 lo
    g1[3] = (int)(((nrows >> 16) & 0xFFFFu) | ((width_elems & 0xFFFFu) << 16));   // dim1 hi | tile_dim0
    g1[4] = (int)(nrows & 0xFFFFu);                      // tile_dim1 | tile_dim2=0
    g1[5] = (int)(stride_elems & 0xFFFFFFFFull);         // tensor_dim0_stride lo32
    g1[6] = (int)((stride_elems >> 32) & 0xFFFFull);     // stride hi16 | dim1_stride lo16=0
    g1[7] = 0;
    i32x4 z4 = {};
    i32x8 z8 = {};
    __builtin_amdgcn_tensor_load_to_lds(g0, g1, z4, z4, z8, 0);
}
#else
#define HAVE_TDM 0
#endif

// ---------------- kernels ----------------

// scalar region layout (floats): [0..47]=rr [48..95]=rr_new [96..143]=pAp
// [144..191]=alpha [192..239]=beta [240]=done
__global__ void zero_scalars_kernel(float* sc) {
    sc[threadIdx.x] = 0.f;
}

// K = 1 + sum of incident (positive) edge weights
__global__ void build_K_kernel(const float* __restrict__ Wc,
                               const float* __restrict__ Wr,
                               float* __restrict__ Kd) {
    int base = blockIdx.x * (NTHREADS * 16) + threadIdx.x;
    #pragma unroll
    for (int k = 0; k < 16; ++k) {
        int idx = base + k * NTHREADS;
        int b = idx / PLANE, rem = idx % PLANE;
        int i = rem / W, j = rem % W;
        const float* wcb = Wc + (size_t)b * (H - 1) * W;
        const float* wrb = Wr + (size_t)b * H * (W - 1);
        float kv = 1.f;
        if (i > 0)     kv += wcb[(i - 1) * W + j];
        if (i < H - 1) kv += wcb[i * W + j];
        if (j > 0)     kv += wrb[i * (W - 1) + j - 1];
        if (j < W - 1) kv += wrb[i * (W - 1) + j];
        Kd[idx] = kv;
    }
}

// r = B - A*B ; p = r ; x = B ; rr[plane] += sum(r*r)
__global__ void init_r_kernel(const float* __restrict__ B,
                              const float* __restrict__ Wc,
                              const float* __restrict__ Wr,
                              const float* __restrict__ Kd,
                              float* __restrict__ x,
                              float* __restrict__ r,
                              float* __restrict__ p,
                              float* sc) {
    __shared__ float red[NTHREADS / 32];
    int plane = blockIdx.x >> 6, chunk = blockIdx.x & 63;
    int b = plane / NC;
    const float* Bp  = B + (size_t)plane * PLANE;
    const float* wcb = Wc + (size_t)b * (H - 1) * W;
    const float* wrb = Wr + (size_t)b * H * (W - 1);
    const float* Kb  = Kd + (size_t)b * PLANE;
    size_t pbase = (size_t)plane * PLANE;
    int t = threadIdx.x;
    float s = 0.f;
    #pragma unroll
    for (int k = 0; k < 16; ++k) {
        int ci = chunk * 4096 + t + k * NTHREADS;
        int i = ci / W, j = ci % W;
        float c  = Bp[ci];
        float up = (i > 0)     ? Bp[ci - W] : 0.f;
        float dn = (i < H - 1) ? Bp[ci + W] : 0.f;
        float lf = (j > 0)     ? Bp[ci - 1] : 0.f;
        float rt = (j < W - 1) ? Bp[ci + 1] : 0.f;
        float wcu = (i > 0)     ? wcb[(i - 1) * W + j] : 0.f;
        float wcd = (i < H - 1) ? wcb[i * W + j]       : 0.f;
        float wrl = (j > 0)     ? wrb[i * (W - 1) + j - 1] : 0.f;
        float wrr = (j < W - 1) ? wrb[i * (W - 1) + j]     : 0.f;
        float y  = Kb[ci] * c - wcu * up - wcd * dn - wrl * lf - wrr * rt;
        float rv = c - y;
        x[pbase + ci] = c;
        r[pbase + ci] = rv;
        p[pbase + ci] = rv;
        s += rv * rv;
    }
    float tot = blockReduceSum(s, red);
    if (t == 0) atomicAdd(&sc[0 + plane], tot);
}

// Fused: p_new = r + beta*p_in (TDM slabs in LDS, incl. halos) ;
//        Ap = A*p_new ; p_out = p_new ; pAp[plane] += p_new . Ap
__global__ void stencil_fused_kernel(const float* __restrict__ Wc,
                                     const float* __restrict__ Wr,
                                     const float* __restrict__ Kd,
                                     const float* __restrict__ r,
                                     const float* __restrict__ p_in,
                                     float* __restrict__ p_out,
                                     float* __restrict__ Ap,
                                     float* sc) {
    const float* done = sc + 240;
    if (*done != 0.f) return;

    __shared__ float sp[(TILE_H + 2) * W];
    __shared__ float sr[(TILE_H + 2) * W];
    __shared__ float red[NTHREADS / 32];

    int plane = blockIdx.x >> 6;           // 0..47
    int tile  = blockIdx.x & 63;
    int b = plane / NC;
    int row0 = tile * TILE_H;
    int t = threadIdx.x;

    int top = (row0 == 0) ? 1 : 0;
    int bot = (row0 + TILE_H == H) ? 1 : 0;
    int first = row0 - 1 + top;            // first source row
    unsigned nrows = (unsigned)(TILE_H + 2 - top - bot);
    size_t pbase = (size_t)plane * PLANE;
    const float* psrc = p_in + pbase + (size_t)first * W;
    const float* rsrc = r    + pbase + (size_t)first * W;

#if HAVE_TDM
    // Wave 0 only (uniform scalar branch): TDM issues regardless of EXEC.
    if (__builtin_amdgcn_readfirstlane((int)threadIdx.x) == 0) {
        unsigned lds_p = (unsigned)(uintptr_t)&sp[top * W];
        unsigned lds_r = (unsigned)(uintptr_t)&sr[top * W];
        tdm_load_2d_f32(psrc, lds_p, W, nrows, W);
        tdm_load_2d_f32(rsrc, lds_r, W, nrows, W);
        __builtin_amdgcn_s_wait_tensorcnt(0);
    }
#else
    for (unsigned k = t; k < nrows * W; k += NTHREADS) {
        sp[top * W + k] = psrc[k];
        sr[top * W + k] = rsrc[k];
    }
#endif
    // zero halo rows at plane edges (matches jnp.pad zero semantics)
    if (top)  { sp[t] = 0.f; sp[t + NTHREADS] = 0.f;
                sr[t] = 0.f; sr[t + NTHREADS] = 0.f; }
    if (bot)  { int o = (TILE_H + 1) * W;
                sp[o + t] = 0.f; sp[o + t + NTHREADS] = 0.f;
                sr[o + t] = 0.f; sr[o + t + NTHREADS] = 0.f; }
    __syncthreads();

    // p_new = r + beta * p  (halo rows included, consistent with neighbors)
    float bt = sc[192 + plane];
    for (int k = t; k < (TILE_H + 2) * W; k += NTHREADS)
        sp[k] = sr[k] + bt * sp[k];
    __syncthreads();

    const float* wcb = Wc + (size_t)b * (H - 1) * W;
    const float* wrb = Wr + (size_t)b * H * (W - 1);
    const float* Kb  = Kd + (size_t)b * PLANE;
    float dot = 0.f;
    #pragma unroll
    for (int rl = 0; rl < TILE_H; ++rl) {
        int i = row0 + rl, li = rl + 1;
        #pragma unroll
        for (int half = 0; half < 2; ++half) {
            int j = t + half * NTHREADS;
            float c  = sp[li * W + j];
            float up = sp[(li - 1) * W + j];
            float dn = sp[(li + 1) * W + j];
            float lf = (j > 0)     ? sp[li * W + j - 1] : 0.f;
            float rt = (j < W - 1) ? sp[li * W + j + 1] : 0.f;
            float wcu = (i > 0)     ? wcb[(i - 1) * W + j] : 0.f;
            float wcd = (i < H - 1) ? wcb[i * W + j]       : 0.f;
            float wrl = (j > 0)     ? wrb[i * (W - 1) + j - 1] : 0.f;
            float wrr = (j < W - 1) ? wrb[i * (W - 1) + j]     : 0.f;
            float y = Kb[i * W + j] * c - wcu * up - wcd * dn - wrl * lf - wrr * rt;
            Ap[pbase + i * W + j]    = y;
            p_out[pbase + i * W + j] = c;
            dot += c * y;
        }
    }
    float tot = blockReduceSum(dot, red);
    if (t == 0) atomicAdd(&sc[96 + plane], tot);
}

__global__ void alpha_kernel(float* sc) {
    if (sc[240] != 0.f) return;
    int t = threadIdx.x;
    if (t < NPL) sc[144 + t] = nan_to_num_dev(sc[0 + t] / sc[96 + t]);
}

// x += a*p ; r -= a*Ap ; rr_new[plane] += sum(r_new^2)
__global__ void update_xr_kernel(float* __restrict__ x,
                                 const float* __restrict__ p,
                                 const float* __restrict__ Ap,
                                 float* __restrict__ r,
                                 float* sc) {
    if (sc[240] != 0.f) return;
    __shared__ float red[NTHREADS / 32];
    int plane = blockIdx.x >> 6, chunk = blockIdx.x & 63;
    size_t base = (size_t)plane * PLANE + (size_t)chunk * 4096;
    float a = sc[144 + plane];
    int t = threadIdx.x;
    float s = 0.f;
    #pragma unroll
    for (int k = 0; k < 16; ++k) {
        size_t gi = base + t + k * NTHREADS;
        x[gi] = x[gi] + a * p[gi];
        float rv = r[gi] - a * Ap[gi];
        r[gi] = rv;
        s += rv * rv;
    }
    float tot = blockReduceSum(s, red);
    if (t == 0) atomicAdd(&sc[48 + plane], tot);
}

__global__ void beta_kernel(float* sc) {
    if (sc[240] != 0.f) return;
    __shared__ float sh[64];
    int t = threadIdx.x;
    sh[t] = (t < NPL) ? sc[48 + t] : 0.f;
    __syncthreads();
    if (t == 0) {
        float s = 0.f;
        for (int i = 0; i < 64; ++i) s += sh[i];
        if (s < EPS_CONV) sc[240] = 1.f;   // done
    }
    if (t < NPL) {
        float rrn = sc[48 + t];
        sc[192 + t] = nan_to_num_dev(rrn / sc[0 + t]);  // beta
        sc[0 + t]  = rrn;                               // rr <- rr_new
        sc[48 + t] = 0.f;                               // reset rr_new acc
        sc[96 + t] = 0.f;                               // reset pAp acc
    }
}

// ---------------- launch ----------------

extern "C" void kernel_launch(void* const* d_in, const int* in_sizes, int n_in,
                              void* d_out, int out_size, void* d_ws, size_t ws_size,
                              hipStream_t stream) {
    (void)in_sizes; (void)n_in; (void)out_size; (void)ws_size;
    const float* B  = (const float*)d_in[0];   // [16,3,512,512]
    const float* Wc = (const float*)d_in[1];   // [16,511,512]
    const float* Wr = (const float*)d_in[2];   // [16,512,511]
    float* x  = (float*)d_out;                 // x lives in d_out

    const size_t N  = (size_t)NPL * PLANE;     // 12,582,912
    const size_t KN = (size_t)NB * PLANE;      // 4,194,304
    float* ws = (float*)d_ws;
    float* r   = ws;
    float* pa  = ws + N;
    float* pb  = ws + 2 * N;
    float* Ap  = ws + 3 * N;
    float* Kd  = ws + 4 * N;
    float* sc  = ws + 4 * N + KN;              // 256 scalar floats

    zero_scalars_kernel<<<1, 256, 0, stream>>>(sc);
    build_K_kernel<<<(int)(KN / 4096), NTHREADS, 0, stream>>>(Wc, Wr, Kd);
    init_r_kernel<<<GRID1, NTHREADS, 0, stream>>>(B, Wc, Wr, Kd, x, r, pa, sc);

    float* pin = pa;
    float* pout = pb;
    for (int it = 0; it < MAX_ITERS; ++it) {
        stencil_fused_kernel<<<GRID1, NTHREADS, 0, stream>>>(Wc, Wr, Kd, r, pin, pout, Ap, sc);
        alpha_kernel<<<1, 64, 0, stream>>>(sc);
        update_xr_kernel<<<GRID1, NTHREADS, 0, stream>>>(x, pout, Ap, r, sc);
        beta_kernel<<<1, 64, 0, stream>>>(sc);
        float* tmp = pin; pin = pout; pout = tmp;
    }
}